// Loss_60335700574548
// MI455X (gfx1250) — compile-verified
//
#include <hip/hip_runtime.h>
#include <hip/hip_bf16.h>
#include <stdint.h>

// ---------------- constants -----------------
#define NUM_CLASSES 80
#define F_ALPHA 0.25f
#define NEG_TH 0.4f
#define POS_TH 0.5f
#define G_CNT 32
#define N_IMG 2

// per-level: h*w, cumulative anchor offsets (k-major mapping: tid = k*HW + y*w + x)
// L0 100x128, L1 50x64, L2 25x32, L3 13x16, L4 7x8
#define A_TOTAL 153576          // per image
#define CUM0 115200             // 9*12800
#define CUM1 144000             // +9*3200
#define CUM2 151200             // +9*800
#define CUM3 153072             // +9*208

typedef __attribute__((ext_vector_type(2))) float v2f;
typedef __attribute__((ext_vector_type(8))) float v8f;

// ---------- exact 32-lane f32 sum on the matrix pipe ----------
// A(16x4): lane l's VGPR0 is a unique (m,k) slot; set VGPR1 slot to 0.
// B(4x16) = all ones (layout independent).  D[m,n] = v[m] + v[m+16].
// Per-lane sum of the 8 D VGPRs covers half the rows; xor-16 add finishes.
__device__ __forceinline__ float wave_sum32(float v) {
    v2f a; a[0] = v;    a[1] = 0.0f;
    v2f b; b[0] = 1.0f; b[1] = 1.0f;
    v8f c = {};
    c = __builtin_amdgcn_wmma_f32_16x16x4_f32(false, a, false, b, (short)0, c,
                                              false, false);
    float s = ((c[0] + c[1]) + (c[2] + c[3])) + ((c[4] + c[5]) + (c[6] + c[7]));
    s += __shfl_xor(s, 16, 32);
    return s;
}

// ---------- anchor geometry from flat (per-image) anchor id ----------
__device__ __forceinline__ void anchor_from_tid(int t, int& level, int& k, int& pix,
                                                int& HW, float& ax1, float& ay1,
                                                float& ax2, float& ay2, float& area) {
    int i;
    if (t < CUM0)      { level = 0; i = t;        HW = 12800; }
    else if (t < CUM1) { level = 1; i = t - CUM0; HW = 3200;  }
    else if (t < CUM2) { level = 2; i = t - CUM1; HW = 800;   }
    else if (t < CUM3) { level = 3; i = t - CUM2; HW = 208;   }
    else               { level = 4; i = t - CUM3; HW = 56;    }
    k   = i / HW;
    pix = i - k * HW;
    int w_log2 = 7 - level;                 // w = 128,64,32,16,8
    int x = pix & ((1 << w_log2) - 1);
    int y = pix >> w_log2;
    float strd = (float)(8 << level);

    const float M4[3]  = {4.0f, 5.03968419957949f, 6.34960420787280f}; // 4*2^(j/3)
    const float HRT[3] = {0.70710678118654752f, 1.0f, 1.41421356237309505f};
    int ri = k / 3;
    int si = k - ri * 3;
    float sc = M4[si] * strd;
    float hr = HRT[ri];
    float wa = sc / hr;
    float ha = sc * hr;
    float cx = ((float)x + 0.5f) * strd;
    float cy = ((float)y + 0.5f) * strd;
    ax1 = cx - 0.5f * wa; ay1 = cy - 0.5f * ha;
    ax2 = cx + 0.5f * wa; ay2 = cy + 0.5f * ha;
    area = (ax2 - ax1) * (ay2 - ay1);
}

// shared by both kernels so the equality test is bitwise consistent
__device__ __forceinline__ float iou_af(float ax1, float ay1, float ax2, float ay2,
                                        float aarea, float gx1, float gy1, float gx2,
                                        float gy2, float garea) {
    float ltx = fmaxf(gx1, ax1), lty = fmaxf(gy1, ay1);
    float rbx = fminf(gx2, ax2), rby = fminf(gy2, ay2);
    float iw = fmaxf(rbx - ltx, 0.0f), ih = fmaxf(rby - lty, 0.0f);
    float inter = iw * ih;
    return inter * __builtin_amdgcn_rcpf(garea + aarea - inter + 1e-6f);
}

// ---------------- kernel 0: init workspace ----------------
__global__ void k_init(uint32_t* gmax, float* acc) {
    int i = threadIdx.x;
    if (i < N_IMG * G_CNT) gmax[i] = 0u;
    if (i < 3) acc[i] = 0.0f;
}

// ---------------- kernel 1: per-gt max IoU over all anchors ----------------
__global__ void k_gtmax(const float* __restrict__ gtb, uint32_t* __restrict__ gmax) {
    __shared__ float sx1[G_CNT], sy1[G_CNT], sx2[G_CNT], sy2[G_CNT], sga[G_CNT];
    __shared__ uint32_t smax[G_CNT];
    const int n = blockIdx.y;
    const int tid = blockIdx.x * 256 + threadIdx.x;
    if (threadIdx.x < G_CNT) {
        const float* g = gtb + (size_t)(n * G_CNT + threadIdx.x) * 4;
        float x1 = g[0], y1 = g[1], x2 = g[2], y2 = g[3];
        sx1[threadIdx.x] = x1; sy1[threadIdx.x] = y1;
        sx2[threadIdx.x] = x2; sy2[threadIdx.x] = y2;
        sga[threadIdx.x] = (x2 - x1) * (y2 - y1);
        smax[threadIdx.x] = 0u;
    }
    __syncthreads();

    const bool valid = tid < A_TOTAL;
    int t = valid ? tid : (A_TOTAL - 1);
    int level, k, pix, HW;
    float ax1, ay1, ax2, ay2, aarea;
    anchor_from_tid(t, level, k, pix, HW, ax1, ay1, ax2, ay2, aarea);

    for (int g = 0; g < G_CNT; ++g) {
        float v = iou_af(ax1, ay1, ax2, ay2, aarea, sx1[g], sy1[g], sx2[g], sy2[g], sga[g]);
        if (!valid) v = 0.0f;
        v = fmaxf(v, __shfl_xor(v, 1, 32));
        v = fmaxf(v, __shfl_xor(v, 2, 32));
        v = fmaxf(v, __shfl_xor(v, 4, 32));
        v = fmaxf(v, __shfl_xor(v, 8, 32));
        v = fmaxf(v, __shfl_xor(v, 16, 32));
        if ((threadIdx.x & 31) == 0) atomicMax(&smax[g], __float_as_uint(v));
    }
    __syncthreads();
    if (threadIdx.x < G_CNT) atomicMax(&gmax[n * G_CNT + threadIdx.x], smax[threadIdx.x]);
}

// ---------------- kernel 2: assignment + focal + reg loss ----------------
__global__ void k_main(const float* __restrict__ c0, const float* __restrict__ c1,
                       const float* __restrict__ c2, const float* __restrict__ c3,
                       const float* __restrict__ c4, const float* __restrict__ r0,
                       const float* __restrict__ r1, const float* __restrict__ r2,
                       const float* __restrict__ r3, const float* __restrict__ r4,
                       const float* __restrict__ gtb, const int* __restrict__ gtl,
                       const uint32_t* __restrict__ gmax, float* __restrict__ acc) {
    __shared__ float sx1[G_CNT], sy1[G_CNT], sx2[G_CNT], sy2[G_CNT], sga[G_CNT];
    __shared__ float sgm[G_CNT];
    __shared__ int slbl[G_CNT];
    __shared__ float s_acc[3];
    const int n = blockIdx.y;
    const int tid = blockIdx.x * 256 + threadIdx.x;
    if (threadIdx.x < G_CNT) {
        const float* g = gtb + (size_t)(n * G_CNT + threadIdx.x) * 4;
        float x1 = g[0], y1 = g[1], x2 = g[2], y2 = g[3];
        sx1[threadIdx.x] = x1; sy1[threadIdx.x] = y1;
        sx2[threadIdx.x] = x2; sy2[threadIdx.x] = y2;
        sga[threadIdx.x] = (x2 - x1) * (y2 - y1);
        slbl[threadIdx.x] = gtl[n * G_CNT + threadIdx.x];
        sgm[threadIdx.x] = __uint_as_float(gmax[n * G_CNT + threadIdx.x]);
    }
    if (threadIdx.x < 3) s_acc[threadIdx.x] = 0.0f;
    __syncthreads();

    const bool valid = tid < A_TOTAL;
    int t = valid ? tid : (A_TOTAL - 1);
    int level, k, pix, HW;
    float ax1, ay1, ax2, ay2, aarea;
    anchor_from_tid(t, level, k, pix, HW, ax1, ay1, ax2, ay2, aarea);

    const float* cls = (level == 0) ? c0 : (level == 1) ? c1 : (level == 2) ? c2
                                        : (level == 3) ? c3 : c4;
    const float* reg = (level == 0) ? r0 : (level == 1) ? r1 : (level == 2) ? r2
                                        : (level == 3) ? r3 : r4;
    const float* cp = cls + (size_t)(n * 9 * NUM_CLASSES + k * NUM_CLASSES) * (size_t)HW + pix;
    const float* rp = reg + (size_t)(n * 36 + k * 4) * (size_t)HW + pix;

    // ----- assignment -----
    float best = -1.0f;
    int bestg = 0, lastj = -1;
    #pragma unroll
    for (int g = 0; g < G_CNT; ++g) {
        float v = iou_af(ax1, ay1, ax2, ay2, aarea, sx1[g], sy1[g], sx2[g], sy2[g], sga[g]);
        if (v > best) { best = v; bestg = g; }   // first argmax on ties
        if (v == sgm[g]) lastj = g;              // last gt achieving its row-max
    }
    int assigned = -1;
    if (best < NEG_TH) assigned = 0;
    if (best >= POS_TH) assigned = bestg + 1;
    if (lastj >= 0) assigned = lastj + 1;
    const bool pos = valid && (assigned > 0);
    const bool inc = valid && (assigned >= 0);
    const int gi = (assigned > 0) ? (assigned - 1) : 0;
    const int label = pos ? slbl[gi] : NUM_CLASSES;

    // ----- focal loss: sum_c p^2 log(1-p), plus single correction at c==label -----
    float sum2 = 0.0f, t1a = 0.0f, t2a = 0.0f;
    const size_t cstride = (size_t)HW;
    #pragma unroll 8
    for (int c = 0; c < NUM_CLASSES; ++c) {
        if (c + 24 < NUM_CLASSES) __builtin_prefetch(cp + (size_t)(c + 24) * cstride, 0, 0);
        float xg = cp[(size_t)c * cstride];
        float p = __builtin_amdgcn_rcpf(1.0f + __expf(-xg));
        p = fminf(fmaxf(p, 1e-6f), 1.0f - 1e-6f);
        float omp = 1.0f - p;
        float t2 = p * p * __logf(omp);
        sum2 += t2;
        if (c == label) { t1a = omp * omp * __logf(p); t2a = t2; }
    }
    float clsv = inc ? ((1.0f - F_ALPHA) * (t2a - sum2) - F_ALPHA * t1a) : 0.0f;

    // ----- reg L1 vs encoded target -----
    float rg0 = rp[0], rg1 = rp[cstride], rg2 = rp[2 * cstride], rg3 = rp[3 * cstride];
    float gx1 = sx1[gi], gy1 = sy1[gi], gx2 = sx2[gi], gy2 = sy2[gi];
    float aw = ax2 - ax1, ah = ay2 - ay1;
    float axc = (ax1 + ax2) * 0.5f, ayc = (ay1 + ay2) * 0.5f;
    float gw = gx2 - gx1, gh = gy2 - gy1;
    float gxc = (gx1 + gx2) * 0.5f, gyc = (gy1 + gy2) * 0.5f;
    float tx = (gxc - axc) / aw;
    float ty = (gyc - ayc) / ah;
    float tw = __logf(gw / aw);
    float th = __logf(gh / ah);
    float regv = fabsf(rg0 - tx) + fabsf(rg1 - ty) + fabsf(rg2 - tw) + fabsf(rg3 - th);
    regv = pos ? regv : 0.0f;
    float npv = pos ? 1.0f : 0.0f;

    // ----- reduction: WMMA wave sum -> LDS -> global (EXEC is all-ones here) -----
    float cr = wave_sum32(clsv);
    float rr = wave_sum32(regv);
    float nr = wave_sum32(npv);
    if ((threadIdx.x & 31) == 0) {
        atomicAdd(&s_acc[0], cr);
        atomicAdd(&s_acc[1], rr);
        atomicAdd(&s_acc[2], nr);
    }
    __syncthreads();
    if (threadIdx.x == 0) {
        atomicAdd(&acc[0], s_acc[0]);
        atomicAdd(&acc[1], s_acc[1]);
        atomicAdd(&acc[2], s_acc[2]);
    }
}

// ---------------- kernel 3: finalize ----------------
__global__ void k_final(const float* __restrict__ acc, float* __restrict__ out) {
    if (threadIdx.x == 0) {
        float np = fmaxf(acc[2], 1.0f);
        out[0] = acc[0] / np;
        out[1] = acc[1] / np;
    }
}

extern "C" void kernel_launch(void* const* d_in, const int* in_sizes, int n_in,
                              void* d_out, int out_size, void* d_ws, size_t ws_size,
                              hipStream_t stream) {
    // setup_inputs dict order: cls_p0, reg_p0, cls_p1, reg_p1, ..., gt_bboxes, gt_labels
    const float* c[5];
    const float* r[5];
    for (int i = 0; i < 5; ++i) {
        c[i] = (const float*)d_in[2 * i + 0];
        r[i] = (const float*)d_in[2 * i + 1];
    }
    const float* gtb = (const float*)d_in[10];
    const int* gtl = (const int*)d_in[11];

    uint32_t* gmax = (uint32_t*)d_ws;                  // 64 uints (per-image per-gt max bits)
    float* acc = (float*)((char*)d_ws + 256);          // [cls_sum, reg_sum, npos]
    float* out = (float*)d_out;

    dim3 grid((A_TOTAL + 255) / 256, N_IMG);
    k_init<<<dim3(1), dim3(64), 0, stream>>>(gmax, acc);
    k_gtmax<<<grid, dim3(256), 0, stream>>>(gtb, gmax);
    k_main<<<grid, dim3(256), 0, stream>>>(c[0], c[1], c[2], c[3], c[4],
                                           r[0], r[1], r[2], r[3], r[4],
                                           gtb, gtl, gmax, acc);
    k_final<<<dim3(1), dim3(32), 0, stream>>>(acc, out);
}